// Eq_NLMP2_18013092840058
// MI455X (gfx1250) — compile-verified
//
#include <hip/hip_runtime.h>
#include <math.h>
#include <utility>

// ---------------------------------------------------------------------------
// CDNA5 (gfx1250) implementation.  fp32 WMMA (V_WMMA_F32_16X16X4_F32) for the
// edge-shared MLP GEMMs: (16 edges x 10..16) @ (16 x 576/144).  wave32: one
// wave = one workgroup = one 16-edge WMMA row tile (barriers elide to wave
// scope via __launch_bounds__(32)).  All tile loops are template-unrolled so
// every tensor-product index is a compile-time constant (no scratch spills).
// ---------------------------------------------------------------------------

typedef __attribute__((ext_vector_type(2))) float v2f;
typedef __attribute__((ext_vector_type(8))) float v8f;

#define EPW 16  // edges per wave (WMMA M dim)

// ---- compile-time for loop ------------------------------------------------
template <typename F, int... Is>
__device__ __forceinline__ void sfor_impl(F f,
                                          std::integer_sequence<int, Is...>) {
  (f(std::integral_constant<int, Is>{}), ...);
}
template <int N, typename F>
__device__ __forceinline__ void sfor(F f) {
  sfor_impl(f, std::make_integer_sequence<int, N>{});
}

__device__ __forceinline__ v8f wmma4(v2f a, v2f b, v8f c) {
  // D(16x16,f32) = A(16x4,f32) * B(4x16,f32) + C
  return __builtin_amdgcn_wmma_f32_16x16x4_f32(
      /*neg_a=*/false, a, /*neg_b=*/false, b,
      /*c_mod=*/(short)0, c, /*reuse_a=*/false, /*reuse_b=*/false);
}

__global__ __launch_bounds__(32) void edge_kernel(
    const float* __restrict__ hn, const float* __restrict__ he,
    const int* __restrict__ esrc, const int* __restrict__ edst,
    const float* __restrict__ evec, const float* __restrict__ emb,
    const float* __restrict__ wnorm,
    const float* __restrict__ fcW1, const float* __restrict__ fcW2,
    const float* __restrict__ fc2W1, const float* __restrict__ fc2W2,
    float* __restrict__ he_out, float* __restrict__ node_acc, int E) {
  __shared__ float st[16 * 16];

  const int lane = threadIdx.x & 31;
  const int m    = lane & 15;   // A/C row (edge) for this lane; also B/C col n
  const int lh   = lane >> 4;   // lane half
  const int kb   = lh * 2;      // K sub-offset within a K=4 chunk
  const int n    = m;
  const long e0  = (long)blockIdx.x * EPW;
  const bool eok = (e0 + m) < (long)E;
  const int  e   = eok ? (int)e0 + m : (E - 1);  // clamped edge index

  // ---- A fragments from emb (16 edges x 10, padded to K=12) ----------------
  // Unconditional loads from clamped addresses, then select-to-zero (no
  // divergent branches around the loads).
  v2f a1[3];
  {
    const long eb = (long)e * 10;
    sfor<3>([&](auto cc) {
      constexpr int c = decltype(cc)::value;
      const int c0 = 4 * c + kb;
      const float x = emb[eb + (c0 < 10 ? c0 : 0)];
      const float y = emb[eb + (c0 + 1 < 10 ? c0 + 1 : 0)];
      a1[c].x = (eok && c0 < 10) ? x : 0.f;
      a1[c].y = (eok && c0 + 1 < 10) ? y : 0.f;
    });
  }

  // ---- layer 1 of both MLPs: h = relu(emb @ fcW1 / sqrt(10)) ---------------
  const float is10 = 0.3162277660168379f;  // 1/sqrt(10)
  v2f a2[4], a2b[4];
  {
    v8f c1 = {0.f, 0.f, 0.f, 0.f, 0.f, 0.f, 0.f, 0.f};
    sfor<3>([&](auto cc) {
      constexpr int c = decltype(cc)::value;
      const int r0 = 4 * c + kb;
      const float bx = fcW1[(r0 < 10 ? r0 : 0) * 16 + n];
      const float by = fcW1[(r0 + 1 < 10 ? r0 + 1 : 0) * 16 + n];
      v2f b;
      b.x = (r0 < 10) ? bx : 0.f;
      b.y = (r0 + 1 < 10) ? by : 0.f;
      c1 = wmma4(a1[c], b, c1);
    });
    sfor<8>([&](auto rr) {
      constexpr int r = decltype(rr)::value;
      const float v = c1[r] * is10;
      st[(r + 8 * lh) * 16 + n] = v > 0.f ? v : 0.f;
    });
    __syncthreads();
    sfor<4>([&](auto cc) {
      constexpr int c = decltype(cc)::value;
      a2[c].x = st[m * 16 + 4 * c + kb];
      a2[c].y = st[m * 16 + 4 * c + kb + 1];
    });
    __syncthreads();

    v8f c1b = {0.f, 0.f, 0.f, 0.f, 0.f, 0.f, 0.f, 0.f};
    sfor<3>([&](auto cc) {
      constexpr int c = decltype(cc)::value;
      const int r0 = 4 * c + kb;
      const float bx = fc2W1[(r0 < 10 ? r0 : 0) * 16 + n];
      const float by = fc2W1[(r0 + 1 < 10 ? r0 + 1 : 0) * 16 + n];
      v2f b;
      b.x = (r0 < 10) ? bx : 0.f;
      b.y = (r0 + 1 < 10) ? by : 0.f;
      c1b = wmma4(a1[c], b, c1b);
    });
    sfor<8>([&](auto rr) {
      constexpr int r = decltype(rr)::value;
      const float v = c1b[r] * is10;
      st[(r + 8 * lh) * 16 + n] = v > 0.f ? v : 0.f;
    });
    __syncthreads();
    sfor<4>([&](auto cc) {
      constexpr int c = decltype(cc)::value;
      a2b[c].x = st[m * 16 + 4 * c + kb];
      a2b[c].y = st[m * 16 + 4 * c + kb + 1];
    });
    __syncthreads();
  }

  // ---- per-edge state (lanes 0..15 own one edge each) ----------------------
  float s_in[24] = {};      // [he_s | hn_s[src] | hn_s[dst]]
  float vin[12][3] = {};    // [he_v | hn_v[src] | hn_v[dst]]
  float sh1[3] = {};
  float ssc[24] = {}, ddc[12] = {}, vvc[12][3] = {};  // pre-scaled inputs
  int   idn = 0;            // edge_dst

  if (lh == 0 && eok) {
    const float* hep = he + (long)e * 20;
    sfor<8>([&](auto ii) { s_in[decltype(ii)::value] = hep[decltype(ii)::value]; });
    sfor<12>([&](auto ii) {
      constexpr int q = decltype(ii)::value;
      vin[q / 3][q % 3] = hep[8 + q];
    });
    const int is = esrc[e];
    idn = edst[e];
    const float* hs = hn + (long)is * 20;
    const float* hd = hn + (long)idn * 20;
    sfor<8>([&](auto ii) {
      constexpr int i = decltype(ii)::value;
      s_in[8 + i] = hs[i];
      s_in[16 + i] = hd[i];
    });
    sfor<12>([&](auto ii) {
      constexpr int q = decltype(ii)::value;
      vin[4 + q / 3][q % 3] = hs[8 + q];
      vin[8 + q / 3][q % 3] = hd[8 + q];
    });
    const float ex = evec[(long)e * 3 + 0];
    const float ey = evec[(long)e * 3 + 1];
    const float ez = evec[(long)e * 3 + 2];
    const float SQ3  = 1.7320508075688772f;
    const float iSQ3 = 0.5773502691896258f;
    const float rn = sqrtf(ex * ex + ey * ey + ez * ez) + 1e-12f;
    sh1[0] = SQ3 * ex / rn; sh1[1] = SQ3 * ey / rn; sh1[2] = SQ3 * ez / rn;
    const float cs1 = 0.1443375672974065f;  // 1/sqrt(2*24)
    const float cv1 = 0.2041241452319315f;  // 1/sqrt(2*12)
    sfor<24>([&](auto ii) {
      constexpr int i = decltype(ii)::value;
      ssc[i] = s_in[i] * cs1;
    });
    sfor<12>([&](auto ii) {
      constexpr int i = decltype(ii)::value;
      const float d =
          (vin[i][0] * sh1[0] + vin[i][1] * sh1[1] + vin[i][2] * sh1[2]) * iSQ3;
      ddc[i] = d * cv1;
      vvc[i][0] = vin[i][0] * cv1;
      vvc[i][1] = vin[i][1] * cv1;
      vvc[i][2] = vin[i][2] * cv1;
    });
  }

  // ---- GEMM w = h @ (fcW2/4): 36 N-tiles, consumed tile-by-tile ------------
  float out0[12] = {}, svA[4] = {}, o1[4][3] = {};
  sfor<36>([&](auto tt) {
    constexpr int t = decltype(tt)::value;
    v8f cw = {0.f, 0.f, 0.f, 0.f, 0.f, 0.f, 0.f, 0.f};
    sfor<4>([&](auto cc) {
      constexpr int c = decltype(cc)::value;
      const int r0 = 4 * c + kb;
      v2f b;
      b.x = fcW2[(long)r0 * 576 + t * 16 + n];
      b.y = fcW2[(long)(r0 + 1) * 576 + t * 16 + n];
      cw = wmma4(a2[c], b, cw);
    });
    sfor<8>([&](auto rr) {
      constexpr int r = decltype(rr)::value;
      st[(r + 8 * lh) * 16 + n] = cw[r] * 0.25f;  // fcW2 / sqrt(16)
    });
    __syncthreads();
    if (lh == 0) {
      sfor<16>([&](auto jj) {
        constexpr int j = decltype(jj)::value;
        constexpr int col = t * 16 + j;          // compile-time constant
        const float w = st[m * 16 + j];
        if constexpr (col < 288) {               // w_ss (24x12)
          constexpr int i = col / 12, o = col % 12;
          out0[o] += ssc[i] * w;
        } else if constexpr (col < 432) {        // w_vs (12x12)
          constexpr int q = col - 288, i = q / 12, o = q % 12;
          out0[o] += ddc[i] * w;
        } else if constexpr (col < 528) {        // w_sv (24x4)
          constexpr int q = col - 432, i = q / 4, o = q & 3;
          svA[o] += ssc[i] * w;
        } else {                                 // w_vv (12x4)
          constexpr int q = col - 528, i = q / 4, o = q & 3;
          o1[o][0] += vvc[i][0] * w;
          o1[o][1] += vvc[i][1] * w;
          o1[o][2] += vvc[i][2] * w;
        }
      });
    }
    __syncthreads();
  });

  // ---- finalize tensor product #1, gates, prep tensor product #2 -----------
  float ss2[8] = {}, dd2[4] = {}, vv2[4][3] = {};
  if (lh == 0) {
    const float cs2  = 0.25f;                 // 1/sqrt(2*8)
    const float cv2  = 0.3535533905932738f;   // 1/sqrt(2*4)
    const float iSQ3 = 0.5773502691896258f;
    float tmpv[4][3];
    sfor<4>([&](auto oo) {
      constexpr int o = decltype(oo)::value;
      float g = out0[8 + o];
      g = g > 0.f ? g : 0.f;                  // gates = relu(g0[:,8:12])
      tmpv[o][0] = (sh1[0] * svA[o] + o1[o][0]) * g;
      tmpv[o][1] = (sh1[1] * svA[o] + o1[o][1]) * g;
      tmpv[o][2] = (sh1[2] * svA[o] + o1[o][2]) * g;
    });
    sfor<8>([&](auto ii) {
      constexpr int i = decltype(ii)::value;
      float t = out0[i];
      t = t > 0.f ? t : 0.f;                  // tmp_s = relu(g0[:,:8])
      ss2[i] = t * cs2;
    });
    sfor<4>([&](auto ii) {
      constexpr int i = decltype(ii)::value;
      const float d =
          (tmpv[i][0] * sh1[0] + tmpv[i][1] * sh1[1] + tmpv[i][2] * sh1[2]) *
          iSQ3;
      dd2[i] = d * cv2;
      vv2[i][0] = tmpv[i][0] * cv2;
      vv2[i][1] = tmpv[i][1] * cv2;
      vv2[i][2] = tmpv[i][2] * cv2;
    });
  }

  // ---- GEMM w2 = h2 @ (fc2W2/4): 9 N-tiles, consumed tile-by-tile ----------
  float d0[8] = {}, sv2[4] = {}, o12[4][3] = {};
  sfor<9>([&](auto tt) {
    constexpr int t = decltype(tt)::value;
    v8f cw = {0.f, 0.f, 0.f, 0.f, 0.f, 0.f, 0.f, 0.f};
    sfor<4>([&](auto cc) {
      constexpr int c = decltype(cc)::value;
      const int r0 = 4 * c + kb;
      v2f b;
      b.x = fc2W2[(long)r0 * 144 + t * 16 + n];
      b.y = fc2W2[(long)(r0 + 1) * 144 + t * 16 + n];
      cw = wmma4(a2b[c], b, cw);
    });
    sfor<8>([&](auto rr) {
      constexpr int r = decltype(rr)::value;
      st[(r + 8 * lh) * 16 + n] = cw[r] * 0.25f;  // fc2W2 / sqrt(16)
    });
    __syncthreads();
    if (lh == 0) {
      sfor<16>([&](auto jj) {
        constexpr int j = decltype(jj)::value;
        constexpr int col = t * 16 + j;
        const float w = st[m * 16 + j];
        if constexpr (col < 64) {                // w2_ss (8x8)
          constexpr int i = col / 8, o = col % 8;
          d0[o] += ss2[i] * w;
        } else if constexpr (col < 96) {         // w2_vs (4x8)
          constexpr int q = col - 64, i = q / 8, o = q % 8;
          d0[o] += dd2[i] * w;
        } else if constexpr (col < 128) {        // w2_sv (8x4)
          constexpr int q = col - 96, i = q / 4, o = q & 3;
          sv2[o] += ss2[i] * w;
        } else {                                 // w2_vv (4x4)
          constexpr int q = col - 128, i = q / 4, o = q & 3;
          o12[o][0] += vv2[i][0] * w;
          o12[o][1] += vv2[i][1] * w;
          o12[o][2] += vv2[i][2] * w;
        }
      });
    }
    __syncthreads();
  });

  // ---- he_new output (streaming, non-temporal) + segment-sum ---------------
  if (lh == 0 && eok) {
    const float nw = wnorm[e];
    float* ho = he_out + (long)e * 20;
    float* na = node_acc + (long)idn * 20;
    sfor<8>([&](auto ii) {
      constexpr int i = decltype(ii)::value;
      const float v = s_in[i] + d0[i];       // he_s + d0
      __builtin_nontemporal_store(v, &ho[i]);
      atomicAdd(&na[i], v * nw);
    });
    sfor<12>([&](auto qq) {
      constexpr int q = decltype(qq)::value;
      constexpr int o = q / 3, c = q % 3;
      const float v = vin[o][c] + (sh1[c] * sv2[o] + o12[o][c]);  // he_v + d1
      __builtin_nontemporal_store(v, &ho[8 + q]);
      atomicAdd(&na[8 + q], v * nw);
    });
  }
}

// ---------------------------------------------------------------------------
__global__ void node_kernel(const float* __restrict__ hn,
                            const float* __restrict__ nf,
                            const float* __restrict__ WgS,
                            const float* __restrict__ WgV,
                            const float* __restrict__ WoS,
                            const float* __restrict__ WoV,
                            float* __restrict__ out, int N) {
  const int nd = blockIdx.x * blockDim.x + threadIdx.x;
  if (nd >= N) return;
  const float* h = hn + (long)nd * 20;
  const float* f = nf + (long)nd * 20;
  float cs[16], cv[8][3];
#pragma unroll
  for (int i = 0; i < 8; ++i) { cs[i] = h[i]; cs[8 + i] = f[i]; }
#pragma unroll
  for (int i = 0; i < 4; ++i)
#pragma unroll
    for (int c = 0; c < 3; ++c) {
      cv[i][c] = h[8 + i * 3 + c];
      cv[4 + i][c] = f[8 + i * 3 + c];
    }
  float gl[12];
#pragma unroll
  for (int o = 0; o < 12; ++o) {
    float a = 0.f;
#pragma unroll
    for (int i = 0; i < 16; ++i) a += cs[i] * WgS[i * 12 + o];
    gl[o] = a * 0.25f;                        // / sqrt(16)
  }
  const float i8 = 0.3535533905932738f;       // 1/sqrt(8)
  float vl[4][3];
#pragma unroll
  for (int o = 0; o < 4; ++o)
#pragma unroll
    for (int c = 0; c < 3; ++c) {
      float a = 0.f;
#pragma unroll
      for (int i = 0; i < 8; ++i) a += cv[i][c] * WgV[i * 4 + o];
      vl[o][c] = a * i8;
    }
  float ls[8], lg[4];
#pragma unroll
  for (int i = 0; i < 8; ++i) ls[i] = gl[i] > 0.f ? gl[i] : 0.f;
#pragma unroll
  for (int o = 0; o < 4; ++o) lg[o] = gl[8 + o] > 0.f ? gl[8 + o] : 0.f;
  float* po = out + (long)nd * 20;
#pragma unroll
  for (int o = 0; o < 8; ++o) {
    float a = 0.f;
#pragma unroll
    for (int i = 0; i < 8; ++i) a += ls[i] * WoS[i * 8 + o];
    po[o] = h[o] + a * i8;                    // / sqrt(8)
  }
#pragma unroll
  for (int o = 0; o < 4; ++o)
#pragma unroll
    for (int c = 0; c < 3; ++c) {
      float a = 0.f;
#pragma unroll
      for (int i = 0; i < 4; ++i) a += vl[i][c] * lg[i] * WoV[i * 4 + o];
      po[8 + o * 3 + c] = h[8 + o * 3 + c] + a * 0.5f;  // / sqrt(4)
    }
}

__global__ void zero_kernel(float* __restrict__ p, long n) {
  const long i = (long)blockIdx.x * blockDim.x + threadIdx.x;
  if (i < n) p[i] = 0.f;
}

// ---------------------------------------------------------------------------
extern "C" void kernel_launch(void* const* d_in, const int* in_sizes, int n_in,
                              void* d_out, int out_size, void* d_ws,
                              size_t ws_size, hipStream_t stream) {
  const float* hn    = (const float*)d_in[0];
  const float* he    = (const float*)d_in[1];
  const int*   esrc  = (const int*)d_in[2];
  const int*   edst  = (const int*)d_in[3];
  const float* evec  = (const float*)d_in[4];
  const float* emb   = (const float*)d_in[5];
  const float* wnorm = (const float*)d_in[6];
  const float* fcW1  = (const float*)d_in[8];
  const float* fcW2  = (const float*)d_in[9];
  const float* fc2W1 = (const float*)d_in[10];
  const float* fc2W2 = (const float*)d_in[11];
  const float* WgS   = (const float*)d_in[12];
  const float* WgV   = (const float*)d_in[13];
  const float* WoS   = (const float*)d_in[14];
  const float* WoV   = (const float*)d_in[15];

  const int E = in_sizes[6];       // norm has shape (E,)
  const int N = in_sizes[0] / 20;  // hn has shape (N,20)

  float* out    = (float*)d_out;
  float* hn_out = out;                   // (N,20), first in return order
  float* he_out = out + (long)N * 20;    // (E,20)
  float* nacc   = (float*)d_ws;          // node accumulator (N,20)

  const long zn = (long)N * 20;
  zero_kernel<<<(int)((zn + 255) / 256), 256, 0, stream>>>(nacc, zn);

  edge_kernel<<<(E + EPW - 1) / EPW, 32, 0, stream>>>(
      hn, he, esrc, edst, evec, emb, wnorm, fcW1, fcW2, fc2W1, fc2W2, he_out,
      nacc, E);

  node_kernel<<<(N + 127) / 128, 128, 0, stream>>>(hn, nacc, WgS, WgV, WoS,
                                                   WoV, hn_out, N);
}